// TransformerBlock_37383395344889
// MI455X (gfx1250) — compile-verified
//
#include <hip/hip_runtime.h>
#include <hip/hip_bf16.h>

// ---------------------------------------------------------------------------
// Transformer block (GQA attention + top-6 MoE of 64 experts + shared SwiGLU)
// for MI455X / gfx1250.  All dense GEMMs use V_WMMA_F32_16X16X4_F32 (full
// fp32 precision; workload is HBM-bound on ~400MB of fp32 expert weights, so
// fp32 WMMA is the right precision/bandwidth point).  MoE is computed
// sparsely, expert-grouped, with a deterministic combine (no float atomics).
// This revision: 16x32 C tiles per wave (2 WMMA accumulators, shared A frag),
// vectorized A loads, and global_prefetch_b8 streaming hints on weight reads.
// ---------------------------------------------------------------------------

#define TT   1024   // tokens (B*S)
#define DD   1024   // model dim
#define NH_  16
#define NKV_ 4
#define HD_  64
#define EE   64     // experts
#define KT   6      // top-k
#define HE_  512

typedef __attribute__((ext_vector_type(2))) float v2f;
typedef __attribute__((ext_vector_type(8))) float v8f;

__device__ __forceinline__ v8f wmma4(v2f a, v2f b, v8f c) {
  // D = A(16x4) * B(4x16) + C(16x16), fp32
  return __builtin_amdgcn_wmma_f32_16x16x4_f32(false, a, false, b, (short)0, c,
                                               false, false);
}

__device__ __forceinline__ float warp_sum(float v) {
  for (int o = 16; o; o >>= 1) v += __shfl_xor(v, o);
  return v;
}
__device__ __forceinline__ float warp_max(float v) {
  for (int o = 16; o; o >>= 1) v = fmaxf(v, __shfl_xor(v, o));
  return v;
}

// ---------------------------------------------------------------- RMSNorm ---
__global__ void rmsnorm_kernel(const float* __restrict__ x,
                               const float* __restrict__ w,
                               float* __restrict__ y, int d) {
  int row = blockIdx.x;
  const float* xr = x + (size_t)row * d;
  float ss = 0.f;
  for (int i = threadIdx.x; i < d; i += blockDim.x) { float v = xr[i]; ss += v * v; }
  __shared__ float red[32];
  ss = warp_sum(ss);
  if ((threadIdx.x & 31) == 0) red[threadIdx.x >> 5] = ss;
  __syncthreads();
  if (threadIdx.x < 32) {
    float t = (threadIdx.x < (blockDim.x >> 5)) ? red[threadIdx.x] : 0.f;
    t = warp_sum(t);
    if (threadIdx.x == 0) red[0] = t;
  }
  __syncthreads();
  float r = rsqrtf(red[0] / (float)d + 1e-5f);
  float* yr = y + (size_t)row * d;
  for (int i = threadIdx.x; i < d; i += blockDim.x) yr[i] = xr[i] * r * w[i];
}

// --------------------------------------------------- generic fp32 WMMA GEMM -
// C[M,N] = A[M,K] @ B[K,N] (+ Res).  One wave per 16x32 C tile (2 WMMA accs).
__global__ void gemm_f32_wmma(const float* __restrict__ A,
                              const float* __restrict__ B,
                              float* __restrict__ C,
                              const float* __restrict__ Res,
                              int M, int N, int Kd) {
  int lane = threadIdx.x & 31;
  int n0 = blockIdx.x * 32, m0 = blockIdx.y * 16;
  int mA = m0 + (lane & 15);
  int kk = (lane >> 4) << 1;
  int nB = n0 + (lane & 15);
  v8f acc0 = {}, acc1 = {};
  const float* ar = A + (size_t)mA * Kd + kk;
  for (int k0 = 0; k0 < Kd; k0 += 4) {
    v2f a = *(const v2f*)(ar + k0);               // contiguous, 8B aligned
    const float* bp = B + (size_t)(k0 + kk) * N + nB;
    v2f b0, b1;
    b0.x = bp[0];      b0.y = bp[N];
    b1.x = bp[16];     b1.y = bp[N + 16];
    __builtin_prefetch(bp + 16 * (size_t)N, 0, 3);  // stream B 16 K-rows ahead
    acc0 = wmma4(a, b0, acc0);
    acc1 = wmma4(a, b1, acc1);
  }
  int rh = (lane >> 4) * 8, col = n0 + (lane & 15);
#pragma unroll
  for (int r = 0; r < 8; ++r) {
    int row = m0 + r + rh;
    float v0 = acc0[r], v1 = acc1[r];
    if (Res) {
      v0 += Res[(size_t)row * N + col];
      v1 += Res[(size_t)row * N + col + 16];
    }
    C[(size_t)row * N + col]      = v0;
    C[(size_t)row * N + col + 16] = v1;
  }
}

// ------------------------------------------- per-head RMSNorm + RoPE (wave) -
__global__ void qknorm_rope_kernel(float* __restrict__ q,
                                   const float* __restrict__ w,
                                   const float* __restrict__ cosb,
                                   const float* __restrict__ sinb, int nH) {
  int wid = (blockIdx.x * blockDim.x + threadIdx.x) >> 5;
  int lane = threadIdx.x & 31;
  int t = wid / nH, hh = wid % nH;
  float* p = q + ((size_t)t * nH + hh) * HD_;
  float x0 = p[2 * lane], x1 = p[2 * lane + 1];
  float ss = warp_sum(x0 * x0 + x1 * x1);
  float r = rsqrtf(ss / (float)HD_ + 1e-5f);
  x0 = x0 * r * w[2 * lane];
  x1 = x1 * r * w[2 * lane + 1];
  float c = cosb[t * (HD_ / 2) + lane], s = sinb[t * (HD_ / 2) + lane];
  p[2 * lane]     = x0 * c - x1 * s;
  p[2 * lane + 1] = x0 * s + x1 * c;
}

// -------------------------------------- flash attention: one wave per (h,q) -
__global__ void attn_kernel(const float* __restrict__ Q,
                            const float* __restrict__ Km,
                            const float* __restrict__ Vm,
                            float* __restrict__ O) {
  int wslot = threadIdx.x >> 5;
  int wid = blockIdx.x * (blockDim.x >> 5) + wslot;
  int lane = threadIdx.x & 31;
  int h = wid / TT, qp = wid % TT;
  __shared__ float qs[8][HD_];
  const float* qr = Q + ((size_t)qp * NH_ + h) * HD_;
  qs[wslot][lane] = qr[lane];
  qs[wslot][lane + 32] = qr[lane + 32];
  __syncthreads();
  int kvh = h / (NH_ / NKV_);
  float m = -INFINITY, l = 0.f, o0 = 0.f, o1 = 0.f;
  for (int j0 = 0; j0 <= qp; j0 += 32) {
    int j = j0 + lane;
    bool valid = (j <= qp);
    float s;
    if (valid) {
      const float* kr = Km + ((size_t)j * NKV_ + kvh) * HD_;
      s = 0.f;
#pragma unroll 8
      for (int d = 0; d < HD_; ++d) s += qs[wslot][d] * kr[d];
      s *= 0.125f;  // 1/sqrt(64)
    } else {
      s = -INFINITY;
    }
    float mnew = fmaxf(m, warp_max(s));
    float scale = __expf(m - mnew);
    float p = valid ? __expf(s - mnew) : 0.f;
    l = l * scale + warp_sum(p);
    o0 *= scale; o1 *= scale;
    for (int jj = 0; jj < 32; ++jj) {
      float pj = __shfl(p, jj);      // uniform per wave
      if (pj > 0.f) {
        const float* vr = Vm + ((size_t)(j0 + jj) * NKV_ + kvh) * HD_;
        o0 += pj * vr[lane];
        o1 += pj * vr[lane + 32];
      }
    }
    m = mnew;
  }
  float inv = 1.f / l;
  float* orow = O + (size_t)qp * (NH_ * HD_) + h * HD_;
  orow[lane] = o0 * inv;
  orow[lane + 32] = o1 * inv;
}

// -------------------------------------------- routing: softmax + top-6 wave -
__global__ void route_kernel(const float* __restrict__ logits,
                             int* __restrict__ cnt, int* __restrict__ tok_list,
                             float* __restrict__ w_list,
                             int* __restrict__ eslot) {
  int t = blockIdx.x * (blockDim.x >> 5) + (threadIdx.x >> 5);
  int lane = threadIdx.x & 31;
  float l0 = logits[t * EE + lane], l1 = logits[t * EE + 32 + lane];
  float mx = warp_max(fmaxf(l0, l1));
  float e0 = __expf(l0 - mx), e1 = __expf(l1 - mx);
  float sum = warp_sum(e0 + e1);
  float p0 = e0 / sum, p1 = e1 / sum;
  float wsel[KT]; int isel[KT];
#pragma unroll
  for (int it = 0; it < KT; ++it) {
    float v; int idx;
    if (p0 >= p1) { v = p0; idx = lane; } else { v = p1; idx = lane + 32; }
    for (int o = 16; o; o >>= 1) {
      float ov = __shfl_xor(v, o); int oi = __shfl_xor(idx, o);
      if (ov > v || (ov == v && oi < idx)) { v = ov; idx = oi; }
    }
    wsel[it] = v; isel[it] = idx;
    if (idx == lane) p0 = -1.f; else if (idx == lane + 32) p1 = -1.f;
  }
  if (lane == 0) {
    float s = 0.f;
#pragma unroll
    for (int it = 0; it < KT; ++it) s += wsel[it];
    float invs = 1.f / s;
#pragma unroll
    for (int it = 0; it < KT; ++it) {
      int e = isel[it];
      int pos = atomicAdd(&cnt[e], 1);
      tok_list[e * TT + pos] = t;
      w_list[e * TT + pos] = wsel[it] * invs;
      eslot[t * KT + it] = (e << 10) | pos;
    }
  }
}

__global__ void zero_counts_kernel(int* cnt) {
  if (threadIdx.x < EE) cnt[threadIdx.x] = 0;
}
__global__ void scan_offsets_kernel(const int* cnt, int* offs) {
  if (threadIdx.x == 0) {
    int acc = 0;
    for (int e = 0; e < EE; ++e) { offs[e] = acc; acc += cnt[e]; }
  }
}

// ---------------- fused dual GEMM + SiLU: G = silu(Z@W1)*(Z@W3)[*w_row] ------
// Gathered (MoE, cnt!=null, blockIdx.z = expert) or dense (cnt==null).
__global__ void swiglu_up_kernel(const float* __restrict__ Z,
                                 const float* __restrict__ W1,
                                 const float* __restrict__ W3,
                                 float* __restrict__ G,
                                 const int* __restrict__ cnt,
                                 const int* __restrict__ offs,
                                 const int* __restrict__ tok_list,
                                 const float* __restrict__ w_list,
                                 int Mdense, int Nn, int Kd) {
  int e = blockIdx.z;
  int n_e, off;
  const int* tl = nullptr; const float* wl = nullptr;
  size_t wstride = 0;
  if (cnt) {
    n_e = cnt[e]; off = offs[e];
    tl = tok_list + e * TT; wl = w_list + e * TT;
    wstride = (size_t)e * Kd * Nn;
  } else { n_e = Mdense; off = 0; }
  int row0 = blockIdx.y * 16;
  if (row0 >= n_e) return;
  int lane = threadIdx.x & 31;
  int n0 = blockIdx.x * 16;
  int mrow = row0 + (lane & 15);
  int mcl = min(mrow, n_e - 1);
  int tok = tl ? tl[mcl] : mcl;
  int kk = (lane >> 4) << 1;
  int nB = n0 + (lane & 15);
  const float* w1p = W1 + wstride;
  const float* w3p = W3 + wstride;
  const float* zr = Z + (size_t)tok * Kd + kk;
  v8f a1 = {}, a3 = {};
  for (int k0 = 0; k0 < Kd; k0 += 4) {
    v2f a = *(const v2f*)(zr + k0);
    size_t bo = (size_t)(k0 + kk) * Nn + nB;
    v2f b1, b3;
    b1.x = w1p[bo]; b1.y = w1p[bo + Nn];
    b3.x = w3p[bo]; b3.y = w3p[bo + Nn];
    __builtin_prefetch(w1p + bo + 16 * (size_t)Nn, 0, 3);
    __builtin_prefetch(w3p + bo + 16 * (size_t)Nn, 0, 3);
    a1 = wmma4(a, b1, a1);
    a3 = wmma4(a, b3, a3);
  }
  int rh = (lane >> 4) * 8, col = n0 + (lane & 15);
#pragma unroll
  for (int r = 0; r < 8; ++r) {
    int row = row0 + r + rh;
    if (row < n_e) {
      float h1 = a1[r], h3 = a3[r];
      float g = h1 / (1.f + __expf(-h1)) * h3;  // silu(h1)*h3
      if (wl) g *= wl[row];
      G[(size_t)(off + row) * Nn + col] = g;
    }
  }
}

// --------- MoE down-proj: yb[assignment_rows, D] = G_e @ W2_e (per expert) --
// One wave per 16x32 tile (2 WMMA accumulators).
__global__ void moe_down_kernel(const float* __restrict__ G,
                                const float* __restrict__ W2,
                                const int* __restrict__ cnt,
                                const int* __restrict__ offs,
                                float* __restrict__ yb, int Nn, int Kd) {
  int e = blockIdx.z;
  int n_e = cnt[e];
  int row0 = blockIdx.y * 16;
  if (row0 >= n_e) return;
  int off = offs[e];
  int lane = threadIdx.x & 31;
  int n0 = blockIdx.x * 32;
  int mcl = min(row0 + (lane & 15), n_e - 1);
  int kk = (lane >> 4) << 1;
  int nB = n0 + (lane & 15);
  const float* gr = G + (size_t)(off + mcl) * Kd + kk;
  const float* w2p = W2 + (size_t)e * Kd * Nn;
  v8f acc0 = {}, acc1 = {};
  for (int k0 = 0; k0 < Kd; k0 += 4) {
    v2f a = *(const v2f*)(gr + k0);
    const float* bp = w2p + (size_t)(k0 + kk) * Nn + nB;
    v2f b0, b1;
    b0.x = bp[0];  b0.y = bp[Nn];
    b1.x = bp[16]; b1.y = bp[Nn + 16];
    __builtin_prefetch(bp + 16 * (size_t)Nn, 0, 3);
    acc0 = wmma4(a, b0, acc0);
    acc1 = wmma4(a, b1, acc1);
  }
  int rh = (lane >> 4) * 8, col = n0 + (lane & 15);
#pragma unroll
  for (int r = 0; r < 8; ++r) {
    int row = row0 + r + rh;
    if (row < n_e) {
      yb[(size_t)(off + row) * Nn + col]      = acc0[r];
      yb[(size_t)(off + row) * Nn + col + 16] = acc1[r];
    }
  }
}

// -------- deterministic combine: out[t] += sum_{it<6} yb[slot(t,it)] --------
__global__ void moe_combine_kernel(const float* __restrict__ yb,
                                   const int* __restrict__ eslot,
                                   const int* __restrict__ offs,
                                   float* __restrict__ out, int d) {
  int t = blockIdx.x;
  int base[KT];
#pragma unroll
  for (int it = 0; it < KT; ++it) {
    int ep = eslot[t * KT + it];
    base[it] = offs[ep >> 10] + (ep & 1023);
  }
  for (int c = threadIdx.x; c < d; c += blockDim.x) {
    float s = out[(size_t)t * d + c];
#pragma unroll
    for (int it = 0; it < KT; ++it) s += yb[(size_t)base[it] * d + c];
    out[(size_t)t * d + c] = s;
  }
}

// ---------------------------------------------------------------------------
extern "C" void kernel_launch(void* const* d_in, const int* in_sizes, int n_in,
                              void* d_out, int out_size, void* d_ws,
                              size_t ws_size, hipStream_t stream) {
  const float* x        = (const float*)d_in[0];
  const float* fcos     = (const float*)d_in[1];
  const float* fsin     = (const float*)d_in[2];
  const float* attn_nw  = (const float*)d_in[3];
  const float* ffn_nw   = (const float*)d_in[4];
  const float* wq       = (const float*)d_in[5];
  const float* wk       = (const float*)d_in[6];
  const float* wv       = (const float*)d_in[7];
  const float* wo       = (const float*)d_in[8];
  const float* q_nw     = (const float*)d_in[9];
  const float* k_nw     = (const float*)d_in[10];
  const float* gate_w   = (const float*)d_in[11];
  const float* w1_e     = (const float*)d_in[12];
  const float* w3_e     = (const float*)d_in[13];
  const float* w2_e     = (const float*)d_in[14];
  const float* sw1      = (const float*)d_in[15];
  const float* sw3      = (const float*)d_in[16];
  const float* sw2      = (const float*)d_in[17];
  float* out = (float*)d_out;

  // ---- workspace carve (~64 MB of fp32 + small int region) ----
  float* ws = (float*)d_ws;
  size_t p = 0;
  auto alloc = [&](size_t n) { float* r = ws + p; p += n; return r; };
  float* hx     = alloc((size_t)TT * DD);
  float* qlin   = alloc((size_t)TT * NH_ * HD_);
  float* klin   = alloc((size_t)TT * NKV_ * HD_);
  float* vlin   = alloc((size_t)TT * NKV_ * HD_);
  float* attno  = alloc((size_t)TT * NH_ * HD_);
  float* hbuf   = alloc((size_t)TT * DD);
  float* zbuf   = alloc((size_t)TT * DD);
  float* logits = alloc((size_t)TT * EE);
  float* sg     = alloc((size_t)TT * HE_);
  float* gbuf   = alloc((size_t)(TT * KT + 16) * HE_);
  float* yb     = alloc((size_t)(TT * KT + 16) * DD);
  float* w_list = alloc((size_t)EE * TT);
  int* ip = (int*)(ws + p);
  int* cnt      = ip;             ip += EE;
  int* offs     = ip;             ip += EE;
  int* tok_list = ip;             ip += EE * TT;
  int* eslot    = ip;             ip += TT * KT;

  // 1) attn RMSNorm
  rmsnorm_kernel<<<TT, 256, 0, stream>>>(x, attn_nw, hx, DD);
  // 2) Q/K/V projections (WMMA)
  gemm_f32_wmma<<<dim3(DD / 32, TT / 16), 32, 0, stream>>>(hx, wq, qlin, nullptr, TT, DD, DD);
  gemm_f32_wmma<<<dim3((NKV_ * HD_) / 32, TT / 16), 32, 0, stream>>>(hx, wk, klin, nullptr, TT, NKV_ * HD_, DD);
  gemm_f32_wmma<<<dim3((NKV_ * HD_) / 32, TT / 16), 32, 0, stream>>>(hx, wv, vlin, nullptr, TT, NKV_ * HD_, DD);
  // 3) q/k norm + RoPE
  qknorm_rope_kernel<<<(TT * NH_) / 8, 256, 0, stream>>>(qlin, q_nw, fcos, fsin, NH_);
  qknorm_rope_kernel<<<(TT * NKV_) / 8, 256, 0, stream>>>(klin, k_nw, fcos, fsin, NKV_);
  // 4) causal GQA flash attention
  attn_kernel<<<(NH_ * TT) / 8, 256, 0, stream>>>(qlin, klin, vlin, attno);
  // 5) output projection + residual
  gemm_f32_wmma<<<dim3(DD / 32, TT / 16), 32, 0, stream>>>(attno, wo, hbuf, x, TT, DD, DD);
  // 6) ffn RMSNorm
  rmsnorm_kernel<<<TT, 256, 0, stream>>>(hbuf, ffn_nw, zbuf, DD);
  // 7) gate logits + routing
  gemm_f32_wmma<<<dim3(EE / 32, TT / 16), 32, 0, stream>>>(zbuf, gate_w, logits, nullptr, TT, EE, DD);
  zero_counts_kernel<<<1, 64, 0, stream>>>(cnt);
  route_kernel<<<TT / 8, 256, 0, stream>>>(logits, cnt, tok_list, w_list, eslot);
  scan_offsets_kernel<<<1, 32, 0, stream>>>(cnt, offs);
  // 8) shared expert: sg = silu(z@sw1)*(z@sw3); out = h + sg@sw2
  swiglu_up_kernel<<<dim3(HE_ / 16, TT / 16, 1), 32, 0, stream>>>(
      zbuf, sw1, sw3, sg, nullptr, nullptr, nullptr, nullptr, TT, HE_, DD);
  gemm_f32_wmma<<<dim3(DD / 32, TT / 16), 32, 0, stream>>>(sg, sw2, out, hbuf, TT, DD, HE_);
  // 9) MoE experts (grouped): up (gathered, weighted), down, combine
  swiglu_up_kernel<<<dim3(HE_ / 16, TT / 16, EE), 32, 0, stream>>>(
      zbuf, w1_e, w3_e, gbuf, cnt, offs, tok_list, w_list, 0, HE_, DD);
  moe_down_kernel<<<dim3(DD / 32, TT / 16, EE), 32, 0, stream>>>(
      gbuf, w2_e, cnt, offs, yb, DD, HE_);
  moe_combine_kernel<<<TT, 256, 0, stream>>>(yb, eslot, offs, out, DD);
}